// Katies_NeuralSolver_15745350107828
// MI455X (gfx1250) — compile-verified
//
#include <hip/hip_runtime.h>

// MI455X / gfx1250 neural solver:
//   2 × ( gather 3 neighbours -> (64x32) ReLU MLP -> (32x8) -> Euler update )
// fp32 WMMA (V_WMMA_F32_16X16X4_F32): exact fp32 vs the JAX reference.
// One wave32 handles 4 consecutive 16-node tiles with W1/W2/bias fragments
// resident in VGPRs. The GEMM K-dimension is permuted (consistently in A and
// B) so every layer-1 A-fragment is a pair of CONSECUTIVE floats from one
// gathered z-row: A operands are used straight out of the load registers,
// eliminating the per-tile v_mov pair-packing seen in the previous build.

typedef __attribute__((ext_vector_type(2))) float v2f;
typedef __attribute__((ext_vector_type(8))) float v8f;

#define NTOT 163840
#define BATCH 8
#define DIM 16
#define HID 32
#define NDYN 8
#define TILES_PER_BATCH (NTOT / 16)           // 10240
#define TOTAL_TILES (BATCH * TILES_PER_BATCH) // 81920
#define WAVES_PER_BLOCK 8
#define THREADS (WAVES_PER_BLOCK * 32)
#define TILES_PER_WAVE 4
#define NBLOCKS (TOTAL_TILES / (WAVES_PER_BLOCK * TILES_PER_WAVE)) // 2560
#define LDS_STRIDE 34   // padded row (floats): conflict-free banks, 8B-aligned pairs
#define LDS_WAVE (16 * LDS_STRIDE)

__global__ __launch_bounds__(THREADS) void neural_solver_step(
    const float* __restrict__ zin,
    float* __restrict__ zout,
    const int* __restrict__ nbl,
    const float* __restrict__ W1,
    const float* __restrict__ b1,
    const float* __restrict__ W2,
    const float* __restrict__ b2)
{
    __shared__ float hshare[WAVES_PER_BLOCK * LDS_WAVE];

    const int tid  = threadIdx.x;
    const int lane = tid & 31;
    const int wave = tid >> 5;
    const int m    = lane & 15;   // node-in-tile for A rows / output column
    const int hi   = lane >> 4;   // lane half: selects K sub-rows 2,3 vs 0,1

    // ---- Weight fragments: loaded ONCE per wave, reused for all tiles. ----
    // K-permutation: chunk kc, component v, lane-half hi -> original W1 row
    //   kc<8 : row = 8*kc     + 4*v + 2*hi        (self / nb1 channels)
    //   kc>=8: row = 8*(kc-8) + 4*v + 2*hi + 1    (nb0 / nb2 channels)
    // (bijection onto rows 0..63; A uses the same permutation below)
    v2f w1f[16][2];
    #pragma unroll
    for (int kc = 0; kc < 16; ++kc) {
        const int base = (kc < 8) ? (8*kc + 2*hi) : (8*(kc-8) + 2*hi + 1);
        #pragma unroll
        for (int t = 0; t < 2; ++t)
            #pragma unroll
            for (int v = 0; v < 2; ++v)
                w1f[kc][t][v] = W1[(base + 4*v) * HID + 16*t + m];
    }

    // W2 B-frags (cols 8..15 zero-padded); layer-2 K is NOT permuted.
    v2f w2f[8];
    #pragma unroll
    for (int kc = 0; kc < 8; ++kc)
        #pragma unroll
        for (int v = 0; v < 2; ++v) {
            float t = W2[(4*kc + 2*hi + v) * NDYN + (m & 7)];
            w2f[kc][v] = (m < NDYN) ? t : 0.0f;
        }

    // Biases fold into accumulator init (every acc VGPR = column lane%16).
    const float bl0 = b1[m];
    const float bl1 = b1[16 + m];
    const float bc2 = (m < NDYN) ? b2[m & 7] : 0.0f;

    float* hbuf = hshare + wave * LDS_WAVE;  // private per-wave slice, no barrier
    const int wid = blockIdx.x * WAVES_PER_BLOCK + wave;

    #pragma unroll 1
    for (int t = 0; t < TILES_PER_WAVE; ++t) {
        const int tile  = wid * TILES_PER_WAVE + t;
        const int b     = tile / TILES_PER_BATCH;
        const int node0 = (tile - b * TILES_PER_BATCH) * 16;
        const size_t zbase = (size_t)b * NTOT * DIM;

        // ---- Gather rows (each lane holds two full 16-float z rows):
        //   lanes 0-15 : rowA = z[self], rowB = z[nb0]
        //   lanes 16-31: rowA = z[nb1],  rowB = z[nb2]
        const int node = node0 + m;
        const int n0i = nbl[node * 3 + 0];
        const int n1i = nbl[node * 3 + 1];
        const int n2i = nbl[node * 3 + 2];
        const int rowAidx = hi ? n1i : node;
        const int rowBidx = hi ? n2i : n0i;

        const v2f* pA = (const v2f*)(zin + zbase + (size_t)rowAidx * DIM);
        const v2f* pB = (const v2f*)(zin + zbase + (size_t)rowBidx * DIM);
        v2f ra2[8], rb2[8];
        #pragma unroll
        for (int i = 0; i < 8; ++i) { ra2[i] = pA[i]; rb2[i] = pB[i]; }

        v8f acc0, acc1;
        #pragma unroll
        for (int i = 0; i < 8; ++i) { acc0[i] = bl0; acc1[i] = bl1; }

        // ---- Layer 1: 16 nodes x 32 hidden, K=64 -> 16 K-chunks x 2 col-tiles.
        // With the K-permutation, A chunk kc is exactly one v2f of a row:
        //   kc<8  -> ra2[kc]   (self on low lanes, nb1 on high lanes)
        //   kc>=8 -> rb2[kc-8] (nb0 on low lanes, nb2 on high lanes)
        #pragma unroll
        for (int kc = 0; kc < 16; ++kc) {
            const v2f A = (kc < 8) ? ra2[kc] : rb2[kc - 8];
            acc0 = __builtin_amdgcn_wmma_f32_16x16x4_f32(false, A, false, w1f[kc][0],
                                                         (short)0, acc0, false, false);
            acc1 = __builtin_amdgcn_wmma_f32_16x16x4_f32(false, A, false, w1f[kc][1],
                                                         (short)0, acc1, false, false);
        }

        // ReLU
        #pragma unroll
        for (int i = 0; i < 8; ++i) {
            acc0[i] = fmaxf(acc0[i], 0.0f);
            acc1[i] = fmaxf(acc1[i], 0.0f);
        }

        // ---- Transpose h (D layout: row=VGPR, col=lane) -> A layout via LDS.
        #pragma unroll
        for (int r = 0; r < 8; ++r) {
            const int nrow = r + hi * 8;                 // node row of this VGPR
            hbuf[nrow * LDS_STRIDE + m]      = acc0[r];  // h[node, col 0..15]
            hbuf[nrow * LDS_STRIDE + 16 + m] = acc1[r];  // h[node, col 16..31]
        }

        v8f acc2;
        #pragma unroll
        for (int i = 0; i < 8; ++i) acc2[i] = bc2;

        // ---- Layer 2: 16 nodes x 8(16) out, K=32 -> 8 K-chunks.
        // A components are consecutive LDS floats (8B-aligned: stride 34).
        #pragma unroll
        for (int kc = 0; kc < 8; ++kc) {
            const int ch = 4*kc + 2*hi;
            v2f A; A[0] = hbuf[m * LDS_STRIDE + ch];
                   A[1] = hbuf[m * LDS_STRIDE + ch + 1];
            acc2 = __builtin_amdgcn_wmma_f32_16x16x4_f32(false, A, false, w2f[kc],
                                                         (short)0, acc2, false, false);
        }

        // ---- Euler update: z_out = z_in (+ 0.1*F on first 8 channels).
        // acc2 VGPR r = node (r + 8*hi), column m. z rows are L1-hot.
        #pragma unroll
        for (int r = 0; r < 8; ++r) {
            const size_t idx = zbase + (size_t)(node0 + r + hi*8) * DIM + m;
            const float zv = zin[idx];
            zout[idx] = (m < NDYN) ? fmaf(0.1f, acc2[r], zv) : zv;
        }
    }
}

extern "C" void kernel_launch(void* const* d_in, const int* in_sizes, int n_in,
                              void* d_out, int out_size, void* d_ws, size_t ws_size,
                              hipStream_t stream) {
    const float* z_old = (const float*)d_in[0];
    const float* W1    = (const float*)d_in[1];
    const float* b1    = (const float*)d_in[2];
    const float* W2    = (const float*)d_in[3];
    const float* b2    = (const float*)d_in[4];
    const int*   nbl   = (const int*)d_in[5];

    float* ztmp = (float*)d_ws;   // needs BATCH*NTOT*DIM*4 = 84 MB scratch
    float* zout = (float*)d_out;

    // Step 2 gathers from updated z -> grid-wide dependency -> two launches.
    neural_solver_step<<<NBLOCKS, THREADS, 0, stream>>>(z_old, ztmp, nbl, W1, b1, W2, b2);
    neural_solver_step<<<NBLOCKS, THREADS, 0, stream>>>(ztmp,  zout, nbl, W1, b1, W2, b2);

    (void)in_sizes; (void)n_in; (void)out_size; (void)ws_size;
}